// QSScan_15994458211405
// MI455X (gfx1250) — compile-verified
//
#include <hip/hip_runtime.h>
#include <hip/hip_bf16.h>
#include <stdint.h>

// QSScan (Mamba-1 selective scan, int8-dequant fused) for MI455X / gfx1250.
//   BATCH=2, DIM=4096, L=2048, N=16.
// Design: 1 thread = 1 (b,d) channel (8192 threads = 256 single-wave blocks).
// B/C chunks staged to LDS with global_load_async_to_lds_b32 (transposed to
// t-major during the per-lane scatter), double buffered, synced with
// s_wait_asynccnt; scales folded into the once-per-chunk int->float pass.
// State x[16] lives in VGPRs as 8 x float2 for v_pk_* packed-f32 math; decays
// via v_exp_f32. u/dt/z streams are software-pipelined; output stored NT.

#define BATCH  2
#define DIM    4096
#define LSEQ   2048
#define NST    16
#define TCH    64                 // timesteps per staged chunk
#define NCHUNK (LSEQ / TCH)       // 32
#define NSUB   (LSEQ / 4)         // 512 int4-subtiles per channel
#define LANES  32                 // wave32: one wave per block

typedef __attribute__((ext_vector_type(2))) float v2f;
typedef __attribute__((ext_vector_type(4))) float v4f;

__device__ __forceinline__ float fexp2f_(float x) { return __builtin_amdgcn_exp2f(x); }
__device__ __forceinline__ float flog2f_(float x) { return __builtin_amdgcn_logf(x); }
__device__ __forceinline__ float frcpf_(float x)  { return __builtin_amdgcn_rcpf(x); }
__device__ __forceinline__ v2f mk2(float a, float b) { v2f r; r.x = a; r.y = b; return r; }

// CDNA5 async copy: global -> LDS, tracked by ASYNCcnt (CDNA5 ISA).
// GVS mode: mem_addr = SADDR(64b) + VADDR(32b); VDST holds LDS byte address.
__device__ __forceinline__ void async_ld_b32(uint32_t lds_off, uint32_t voff, uint64_t sbase) {
  asm volatile("global_load_async_to_lds_b32 %0, %1, %2"
               :: "v"(lds_off), "v"(voff), "s"(sbase) : "memory");
}
__device__ __forceinline__ void wait_async0() {
  asm volatile("s_wait_asynccnt 0" ::: "memory");
}

#define LOG2E 1.44269504088896340736f
#define LN2   0.69314718055994530942f

__global__ __launch_bounds__(LANES) void qsscan_kernel(
    const int* __restrict__ up_,  const int* __restrict__ dtp_,
    const int* __restrict__ Bp_,  const int* __restrict__ Cp_,
    const int* __restrict__ zp_,  const int* __restrict__ Ap_,
    const int* __restrict__ Dp_,  const int* __restrict__ dtbp_,
    const float* __restrict__ su,  const float* __restrict__ sdt,
    const float* __restrict__ sA,  const float* __restrict__ sB,
    const float* __restrict__ sC,  const float* __restrict__ sz,
    const float* __restrict__ sD,  const float* __restrict__ sdtb,
    float* __restrict__ outp_)
{
  // [ Bbuf0 | Bbuf1 | Cbuf0 | Cbuf1 ], each TCH*NST floats, t-major rows of 16.
  __shared__ __align__(16) float smem[4 * TCH * NST];   // 16 KB

  const int lane = threadIdx.x;
  const int ch   = (blockIdx.x << 5) + lane;   // global channel; one batch per block
  const int b    = ch >> 12;                   // / DIM
  const int d    = ch & (DIM - 1);

  const float u_s = *su, dt_s = *sdt, A_s = *sA, B_s = *sB;
  const float C_s = *sC, z_s  = *sz,  D_s = *sD, dtb_s = *sdtb;

  const float dtb = dtb_s * (float)dtbp_[d];
  const float Dd  = D_s  * (float)Dp_[d];

  // A2[n] = log2(e) * A[d][n],  A = -exp(A_log * A_s)  (log2-folded once)
  const int4* Aq = (const int4*)(Ap_ + d * NST);
  int4 a0 = Aq[0], a1 = Aq[1], a2 = Aq[2], a3 = Aq[3];
  int ai[16] = {a0.x,a0.y,a0.z,a0.w, a1.x,a1.y,a1.z,a1.w,
                a2.x,a2.y,a2.z,a2.w, a3.x,a3.y,a3.z,a3.w};
  v2f A2[8], x2[8];
  #pragma unroll
  for (int i = 0; i < 8; ++i) {
    A2[i] = mk2(-LOG2E * fexp2f_(LOG2E * A_s * (float)ai[2*i]),
                -LOG2E * fexp2f_(LOG2E * A_s * (float)ai[2*i+1]));
    x2[i] = mk2(0.f, 0.f);
  }

  const uint64_t gB = (uint64_t)(uintptr_t)(Bp_ + (size_t)b * NST * LSEQ);
  const uint64_t gC = (uint64_t)(uintptr_t)(Cp_ + (size_t)b * NST * LSEQ);

  // Stage one chunk of B and C (16 x TCH ints each) into LDS, transposing
  // n-major (global, rows stride L) -> t-major (LDS rows of 16). 64 async b32
  // per lane per chunk; per-instruction the 32 lanes read contiguous 128B.
  auto stage = [&](int l0, int buf) {
    uint32_t ldsB0 = (uint32_t)(uintptr_t)(&smem[buf << 10]);
    uint32_t ldsC0 = (uint32_t)(uintptr_t)(&smem[(2 << 10) + (buf << 10)]);
    #pragma unroll 4
    for (int i = 0; i < 32; ++i) {
      int q = (i << 5) + lane;                 // 0..1023
      int n = q >> 6;                          // /TCH
      int t = q & (TCH - 1);
      uint32_t go = (uint32_t)(((n << 11) + l0 + t) << 2);   // (n*L + l0+t)*4
      uint32_t lo = (uint32_t)(((t << 4) + n) << 2);         // (t*16 + n)*4
      async_ld_b32(ldsB0 + lo, go, gB);
      async_ld_b32(ldsC0 + lo, go, gC);
    }
  };

  const int4* u4p  = (const int4*)(up_  + (size_t)ch * LSEQ);
  const int4* dt4p = (const int4*)(dtp_ + (size_t)ch * LSEQ);
  const int4* z4p  = (const int4*)(zp_  + (size_t)ch * LSEQ);
  v4f*        o4p  = (v4f*)(outp_ + (size_t)ch * LSEQ);

  stage(0, 0);

  // software-pipelined u/dt/z streams: subtile 0 in flight before chunk 0
  int4 u4 = u4p[0], dt4 = dt4p[0], z4 = z4p[0];

  #pragma unroll 1
  for (int c = 0; c < NCHUNK; ++c) {
    const int buf = c & 1;
    wait_async0();                                   // chunk c landed in LDS
    if (c + 1 < NCHUNK) stage((c + 1) * TCH, (c + 1) & 1);  // overlap next chunk

    float* Bb = &smem[buf << 10];
    float* Cb = &smem[(2 << 10) + (buf << 10)];

    // Once per chunk: int32 -> float in place, folding B_s / C_s scales.
    #pragma unroll
    for (int i = 0; i < 8; ++i) {
      int idx = lane + (i << 5);                     // covers 0..255 float4s
      int4 vb = ((const int4*)Bb)[idx];
      int4 vc = ((const int4*)Cb)[idx];
      ((float4*)Bb)[idx] = make_float4((float)vb.x * B_s, (float)vb.y * B_s,
                                       (float)vb.z * B_s, (float)vb.w * B_s);
      ((float4*)Cb)[idx] = make_float4((float)vc.x * C_s, (float)vc.y * C_s,
                                       (float)vc.z * C_s, (float)vc.w * C_s);
    }

    const int base4 = c * (TCH / 4);
    #pragma unroll 2
    for (int s = 0; s < TCH / 4; ++s) {
      const int g = base4 + s;                       // uniform subtile index
      const int gn = (g + 1 < NSUB) ? g + 1 : g;     // clamped prefetch index
      int4 nu = u4p[gn], ndt = dt4p[gn], nz = z4p[gn];   // prefetch next subtile

      int ua[4]  = {u4.x,  u4.y,  u4.z,  u4.w};
      int dta[4] = {dt4.x, dt4.y, dt4.z, dt4.w};
      int za[4]  = {z4.x,  z4.y,  z4.z,  z4.w};
      float oa[4];
      #pragma unroll
      for (int k = 0; k < 4; ++k) {
        const int t = (s << 2) + k;
        // delta = softplus(dt*dt_s + dt_bias)
        float dv = dt_s * (float)dta[k] + dtb;
        float sp = LN2 * flog2f_(1.f + fexp2f_(LOG2E * dv));
        float delta = (dv > 20.f) ? dv : sp;
        float uf  = u_s * (float)ua[k];
        float duB = delta * uf;                      // B scale already in LDS data

        const float4* Br = (const float4*)(Bb + (t << 4));  // broadcast reads
        const float4* Cr = (const float4*)(Cb + (t << 4));
        float4 B0 = Br[0], B1 = Br[1], B2 = Br[2], B3 = Br[3];
        float4 C0 = Cr[0], C1 = Cr[1], C2 = Cr[2], C3 = Cr[3];
        v2f bv[8] = { mk2(B0.x,B0.y), mk2(B0.z,B0.w), mk2(B1.x,B1.y), mk2(B1.z,B1.w),
                      mk2(B2.x,B2.y), mk2(B2.z,B2.w), mk2(B3.x,B3.y), mk2(B3.z,B3.w) };
        v2f cv[8] = { mk2(C0.x,C0.y), mk2(C0.z,C0.w), mk2(C1.x,C1.y), mk2(C1.z,C1.w),
                      mk2(C2.x,C2.y), mk2(C2.z,C2.w), mk2(C3.x,C3.y), mk2(C3.z,C3.w) };

        v2f d2  = mk2(delta, delta);
        v2f db2 = mk2(duB, duB);
        v2f acc = mk2(0.f, 0.f);
        #pragma unroll
        for (int i = 0; i < 8; ++i) {                // packed-f32 state update
          v2f arg = d2 * A2[i];
          v2f dA  = mk2(fexp2f_(arg.x), fexp2f_(arg.y));
          x2[i] = dA * x2[i] + db2 * bv[i];
          acc   = acc + x2[i] * cv[i];
        }
        float y   = acc.x + acc.y;                   // C scale already in LDS data
        float zf  = z_s * (float)za[k];
        float sig = frcpf_(1.f + fexp2f_(-LOG2E * zf));
        oa[k] = (y + Dd * uf) * (zf * sig);          // (y + D*u) * z * sigmoid(z)
      }
      v4f o; o.x = oa[0]; o.y = oa[1]; o.z = oa[2]; o.w = oa[3];
      __builtin_nontemporal_store(o, &o4p[g]);       // write-once stream: TH=NT
      u4 = nu; dt4 = ndt; z4 = nz;
    }
  }
}

extern "C" void kernel_launch(void* const* d_in, const int* in_sizes, int n_in,
                              void* d_out, int out_size, void* d_ws, size_t ws_size,
                              hipStream_t stream) {
  (void)in_sizes; (void)n_in; (void)out_size; (void)d_ws; (void)ws_size;
  qsscan_kernel<<<dim3((BATCH * DIM) / LANES), dim3(LANES), 0, stream>>>(
      (const int*)d_in[0],  (const int*)d_in[1],  (const int*)d_in[2],
      (const int*)d_in[3],  (const int*)d_in[4],  (const int*)d_in[5],
      (const int*)d_in[6],  (const int*)d_in[7],
      (const float*)d_in[8],  (const float*)d_in[9],  (const float*)d_in[10],
      (const float*)d_in[11], (const float*)d_in[12], (const float*)d_in[13],
      (const float*)d_in[14], (const float*)d_in[15],
      (float*)d_out);
}